// ContinuousAxialPositionalEmbedding_16183436771550
// MI455X (gfx1250) — compile-verified
//
#include <hip/hip_runtime.h>
#include <cstdint>

// Axial positional embedding, MI455X (gfx1250).
// out[a*512 + b, 0:512]    = sin(pos0(a)*w0 + b0)   (f0 row, shared by 512 output rows)
// out[a*512 + b, 512:1024] = sin(pos1(b)*w1 + b1)   (f1 row, shared by 256 output rows)
//
// Roofline: output is 131072 x 1024 x 4B = 512 MiB of pure stores -> ~23 us
// floor at 23.3 TB/s HBM. Unique math is ~393K sines (negligible). So: build
// each unique f0/f1 tile ONCE in LDS, then fan it out to HBM with CDNA5 async
// LDS->global B128 stores (ASYNCcnt-deep, no VGPR staging, NT hint so 512 MB
// of write-once data doesn't churn the 192 MB L2).

#define A_DIM 256
#define B_DIM 512
#define H_DIM 512   // per-axis embedding dim
#define AG    8     // a-rows per block
#define BT    8     // b-rows per block

__global__ __launch_bounds__(256) void axial_emb_async_kernel(
    const float* __restrict__ w0, const float* __restrict__ b0,
    const float* __restrict__ w1, const float* __restrict__ b1,
    const float* __restrict__ pdiv0, const float* __restrict__ pmul0,
    const float* __restrict__ pdiv1, const float* __restrict__ pmul1,
    float* __restrict__ out)
{
    __shared__ float f0s[AG * H_DIM];   // 16 KB: 8 rows of f0
    __shared__ float f1s[BT * H_DIM];   // 16 KB: 8 rows of f1

    const int t     = threadIdx.x;          // 0..255
    const int bbase = blockIdx.x * BT;      // gridDim.x = 512/8 = 64
    const int abase = blockIdx.y * AG;      // gridDim.y = 256/8 = 32

    // One divide per axis (reference: pos/div*mul; identical for div=1).
    const float s0 = pmul0[0] / pdiv0[0];
    const float s1 = pmul1[0] / pdiv1[0];

    // Each thread only ever touches columns t and t+256 -> keep weights in VGPRs.
    const float w0lo = w0[t],       b0lo = b0[t];
    const float w0hi = w0[t + 256], b0hi = b0[t + 256];
    const float w1lo = w1[t],       b1lo = b1[t];
    const float w1hi = w1[t + 256], b1hi = b1[t + 256];

    // ---- unique compute: f0 rows for this a-group (HW v_sin_f32 path) ----
    #pragma unroll
    for (int ag = 0; ag < AG; ++ag) {
        const float p = (float)(abase + ag) * s0;
        f0s[ag * H_DIM + t]       = __sinf(fmaf(p, w0lo, b0lo));
        f0s[ag * H_DIM + t + 256] = __sinf(fmaf(p, w0hi, b0hi));
    }
    // ---- unique compute: f1 rows for this b-chunk ----
    // element index = k*256 + t ; row = k>>1 ; col = t + (k&1)*256
    #pragma unroll
    for (int k = 0; k < 2 * BT; ++k) {
        const int   row = k >> 1;
        const float p   = (float)(bbase + row) * s1;
        const float w   = (k & 1) ? w1hi : w1lo;
        const float bb  = (k & 1) ? b1hi : b1lo;
        f1s[k * 256 + t] = __sinf(fmaf(p, w, bb));
    }
    __syncthreads();   // LDS tiles visible to the async-store engine

    // ---- replication: async LDS -> global streaming stores (16 B per lane) ----
    // Output row = 1024 floats = 4096 B. Threads 0..127 (waves 0-3) stream the
    // f0 half, threads 128..255 (waves 4-7) the f1 half -> wave-uniform select.
    const bool     firstHalf = (t < 128);
    const uint32_t ldsF0 = (uint32_t)(uintptr_t)(&f0s[0]) + (uint32_t)(t * 16);
    const uint32_t ldsF1 = (uint32_t)(uintptr_t)(&f1s[0]) + (uint32_t)((t - 128) * 16);

    const unsigned long long g0 =
        (unsigned long long)(uintptr_t)out
        + ((unsigned long long)(abase * B_DIM + bbase) << 12)   // row * 4096 B
        + (unsigned long long)(t * 16);

    #pragma unroll
    for (int ag = 0; ag < AG; ++ag) {
        #pragma unroll
        for (int row = 0; row < BT; ++row) {
            const unsigned long long ga =
                g0 + ((unsigned long long)(ag * B_DIM + row) << 12);
            const uint32_t lds = firstHalf ? (ldsF0 + (uint32_t)(ag  * 2048))
                                           : (ldsF1 + (uint32_t)(row * 2048));
            // CDNA5 async DMA-style store: LDS -> global, 16B/lane, non-temporal.
            asm volatile("global_store_async_from_lds_b128 %0, %1, off th:TH_STORE_NT"
                         :: "v"(ga), "v"(lds)
                         : "memory");
        }
    }
    // Drain ASYNCcnt before the wave releases its LDS.
    asm volatile("s_wait_asynccnt 0" ::: "memory");
}

extern "C" void kernel_launch(void* const* d_in, const int* in_sizes, int n_in,
                              void* d_out, int out_size, void* d_ws, size_t ws_size,
                              hipStream_t stream) {
    const float* w0   = (const float*)d_in[0];
    const float* b0   = (const float*)d_in[1];
    const float* w1   = (const float*)d_in[2];
    const float* b1   = (const float*)d_in[3];
    const float* div0 = (const float*)d_in[4];
    const float* mul0 = (const float*)d_in[5];
    const float* div1 = (const float*)d_in[6];
    const float* mul1 = (const float*)d_in[7];
    // d_in[8] = seq_len (int), implied by the fixed A/B/H geometry.

    float* out = (float*)d_out;

    dim3 grid(B_DIM / BT, A_DIM / AG);   // (64, 32) = 2048 blocks
    axial_emb_async_kernel<<<grid, 256, 0, stream>>>(
        w0, b0, w1, b1, div0, mul0, div1, mul1, out);
}